// GridMultiAtomLoss_43800076485427
// MI455X (gfx1250) — compile-verified
//
#include <hip/hip_runtime.h>
#include <math.h>

// ---------------------------------------------------------------------------
// GridMultiAtomLoss for MI455X (gfx1250).
// Kernel 1: bandwidth-bound streaming sum of squared diffs over 2 x 83.9 MB.
//           - global_load_b128 with TH_LOAD_NT (single-use data, keep caches clean)
//           - packed v_pk_add/neg + v_pk_mul for the squared diff
//           - accumulation via V_WMMA_F32_16X16X4_F32 with B = ones (row-sums)
//           - uniform unguarded main loop (EXEC all-1s at every WMMA),
//             single guarded tail iteration
//           - deterministic per-block partials in d_ws (no float atomics)
// Kernel 2: tiny finalize (argmin match, CE, BCE) + partial-sum combine.
// ---------------------------------------------------------------------------

typedef __attribute__((ext_vector_type(2))) float v2f;
typedef __attribute__((ext_vector_type(4))) float f32x4;
typedef __attribute__((ext_vector_type(8))) float v8f;

#define MSE_BLOCKS  1024
#define MSE_THREADS 256

__global__ __launch_bounds__(MSE_THREADS) void mse_partials_kernel(
    const float* __restrict__ pred_dist,
    const float* __restrict__ ref_dist,
    float* __restrict__ partials,
    long long n4)
{
    const f32x4* __restrict__ p4 = (const f32x4*)pred_dist;
    const f32x4* __restrict__ r4 = (const f32x4*)ref_dist;

    const long long base   = (long long)blockIdx.x * MSE_THREADS + threadIdx.x;
    const long long stride = (long long)MSE_BLOCKS * MSE_THREADS;

    // base < stride always, so `full` unguarded iterations are in-bounds for
    // every thread and the trip count is grid-uniform (EXEC all-1s at WMMA).
    const long long full = n4 / stride;

    v8f acc = {};
    const v2f ones = {1.0f, 1.0f};

#pragma unroll 4
    for (long long it = 0; it < full; ++it) {
        const long long idx = base + it * stride;
        f32x4 p = __builtin_nontemporal_load(p4 + idx);
        f32x4 r = __builtin_nontemporal_load(r4 + idx);
        float d0 = p.x - r.x;
        float d1 = p.y - r.y;
        float d2 = p.z - r.z;
        float d3 = p.w - r.w;
        // A(16x4) x B(4x16, ones) + C : each distinct A element is summed into
        // its row of D (replicated over 16 columns). 64 f32 reduced per WMMA.
        v2f a0 = {d0 * d0, d1 * d1};
        v2f a1 = {d2 * d2, d3 * d3};
        acc = __builtin_amdgcn_wmma_f32_16x16x4_f32(
            false, a0, false, ones, (short)0, acc, false, false);
        acc = __builtin_amdgcn_wmma_f32_16x16x4_f32(
            false, a1, false, ones, (short)0, acc, false, false);
    }

    // One guarded tail iteration (only the loads are predicated; WMMA executes
    // after reconvergence with zero-filled data for out-of-range lanes).
    {
        const long long idx = base + full * stride;
        f32x4 p = {0.f, 0.f, 0.f, 0.f};
        f32x4 r = {0.f, 0.f, 0.f, 0.f};
        if (idx < n4) {
            p = __builtin_nontemporal_load(p4 + idx);
            r = __builtin_nontemporal_load(r4 + idx);
        }
        float d0 = p.x - r.x;
        float d1 = p.y - r.y;
        float d2 = p.z - r.z;
        float d3 = p.w - r.w;
        v2f a0 = {d0 * d0, d1 * d1};
        v2f a1 = {d2 * d2, d3 * d3};
        acc = __builtin_amdgcn_wmma_f32_16x16x4_f32(
            false, a0, false, ones, (short)0, acc, false, false);
        acc = __builtin_amdgcn_wmma_f32_16x16x4_f32(
            false, a1, false, ones, (short)0, acc, false, false);
    }

    float s = acc[0] + acc[1] + acc[2] + acc[3] + acc[4] + acc[5] + acc[6] + acc[7];

    __shared__ float red[MSE_THREADS];
    red[threadIdx.x] = s;
    __syncthreads();
    for (int off = MSE_THREADS / 2; off > 0; off >>= 1) {
        if (threadIdx.x < off) red[threadIdx.x] += red[threadIdx.x + off];
        __syncthreads();
    }
    // Each row-sum is replicated across 16 columns of D -> divide by 16.
    if (threadIdx.x == 0) partials[blockIdx.x] = red[0] * (1.0f / 16.0f);
}

// ---------------------------------------------------------------------------

#define KB 64   // batch
#define KN 54   // ref nodes
#define KE 54   // edges
#define KC 5    // classes

__global__ __launch_bounds__(256) void finalize_kernel(
    const float* __restrict__ pred_nodes,   // (B, 3+C)
    const float* __restrict__ pred_edges,   // (B, E)
    const float* __restrict__ ref_nodes,    // (B, N, 4)
    const float* __restrict__ ref_edges,    // (B, N, E)
    const int*   __restrict__ ref_terminate,// (B,)
    const float* __restrict__ pred_dist,    // for tail elements (n % 4)
    const float* __restrict__ ref_dist,
    const float* __restrict__ partials,
    int n_partials,
    long long n_total,
    float* __restrict__ out)                // [loss, min_inds(0..63)]
{
    const int tid = threadIdx.x;
    __shared__ float red[256];

    // ---- combine MSE partials (deterministic order) ----
    float a = 0.0f;
    for (int i = tid; i < n_partials; i += 256) a += partials[i];
    if (tid == 0) {
        for (long long i = (n_total / 4) * 4; i < n_total; ++i) {
            float d = pred_dist[i] - ref_dist[i];
            a += d * d;
        }
    }
    red[tid] = a;
    __syncthreads();
    for (int off = 128; off > 0; off >>= 1) {
        if (tid < off) red[tid] += red[tid + off];
        __syncthreads();
    }
    const float mse_mean = red[0] / (float)n_total;  // POS_FACTOR * mean(mse_pos)
    __syncthreads();

    // ---- per-batch matching + CE + BCE (one thread per batch) ----
    float contrib = 0.0f;
    if (tid < KB) {
        const int b = tid;
        const float px = pred_nodes[b * (3 + KC) + 0];
        const float py = pred_nodes[b * (3 + KC) + 1];
        const float pz = pred_nodes[b * (3 + KC) + 2];

        float best = INFINITY;
        int   bi   = 0;
        for (int n = 0; n < KN; ++n) {
            const float* rn = ref_nodes + ((long long)(b * KN + n)) * 4;
            const float dx = rn[0] - px;
            const float dy = rn[1] - py;
            const float dz = rn[2] - pz;
            const float d2 = dx * dx + dy * dy + dz * dz;
            if (d2 < best) { best = d2; bi = n; }   // first-occurrence argmin
        }
        out[1 + b] = (float)bi;

        const int tgt = (int)ref_nodes[((long long)(b * KN + bi)) * 4 + 3];

        // cross entropy: logsumexp(logits) - logits[tgt]
        float l[KC];
        float m = -INFINITY;
        for (int c = 0; c < KC; ++c) {
            l[c] = pred_nodes[b * (3 + KC) + 3 + c];
            m = fmaxf(m, l[c]);
        }
        float se = 0.0f;
        for (int c = 0; c < KC; ++c) se += expf(l[c] - m);
        const float ce = m + logf(se) - l[tgt];

        const float active = (ref_terminate[b] == 0) ? 1.0f : 0.0f;

        // BCE over selected edge row
        float bce = 0.0f;
        const float* sel_row = ref_edges + ((long long)(b * KN + bi)) * KE;
        for (int e = 0; e < KE; ++e) {
            const float p   = pred_edges[b * KE + e];
            const float sel = sel_row[e];
            const float lp  = fmaxf(logf(p), -100.0f);
            const float l1  = fmaxf(log1pf(-p), -100.0f);
            bce += sel * lp + (1.0f - sel) * l1;
        }
        bce = -bce / (float)KE;

        float nll_edge = active * bce;
        if (nll_edge != nll_edge) nll_edge = 0.0f;  // NaN -> 0
        contrib = active * ce + nll_edge;           // CLASS_FACTOR=EDGE_FACTOR=1
    }

    red[tid] = contrib;
    __syncthreads();
    for (int off = 128; off > 0; off >>= 1) {
        if (tid < off) red[tid] += red[tid + off];
        __syncthreads();
    }
    if (tid == 0) out[0] = mse_mean + red[0] * (1.0f / (float)KB);
}

// ---------------------------------------------------------------------------

extern "C" void kernel_launch(void* const* d_in, const int* in_sizes, int n_in,
                              void* d_out, int out_size, void* d_ws, size_t ws_size,
                              hipStream_t stream) {
    const float* pred_nodes    = (const float*)d_in[0];
    const float* pred_edges    = (const float*)d_in[1];
    const float* pred_dist     = (const float*)d_in[2];
    const float* ref_nodes     = (const float*)d_in[3];
    const float* ref_edges     = (const float*)d_in[4];
    const int*   ref_terminate = (const int*)  d_in[5];
    const float* ref_dist      = (const float*)d_in[6];

    float* out      = (float*)d_out;
    float* partials = (float*)d_ws;      // MSE_BLOCKS floats of scratch

    const long long n  = (long long)in_sizes[2];  // 64*20*128*128 = 20,971,520
    const long long n4 = n / 4;

    mse_partials_kernel<<<MSE_BLOCKS, MSE_THREADS, 0, stream>>>(
        pred_dist, ref_dist, partials, n4);

    finalize_kernel<<<1, 256, 0, stream>>>(
        pred_nodes, pred_edges, ref_nodes, ref_edges, ref_terminate,
        pred_dist, ref_dist, partials, MSE_BLOCKS, n, out);
}